// AttentionGNNLayer_90915867721970
// MI455X (gfx1250) — compile-verified
//
#include <hip/hip_runtime.h>
#include <hip/hip_bf16.h>

#define D 128
#define LDSS 132  // padded LDS row stride (floats) to break bank conflicts

typedef float v2f __attribute__((ext_vector_type(2)));
typedef float v8f __attribute__((ext_vector_type(8)));

__device__ __forceinline__ v8f wmma_f32_4(v2f a, v2f b, v8f c) {
  // (neg_a, A, neg_b, B, c_mod, C, reuse_a, reuse_b)
  return __builtin_amdgcn_wmma_f32_16x16x4_f32(false, a, false, b, (short)0, c,
                                               false, false);
}

// ---------------------------------------------------------------------------
// Dual GEMM: Y0 = X @ W0, Y1 = X @ W1, X:(N,128), W:(128,128) row-major.
// Block = 256 threads = 8 waves; wave w owns output columns [16w,16w+16).
// Grid-stride loop over 16-row strips; B fragments loaded once per block and
// held in VGPRs, A tile re-staged in LDS per strip. N divisible by 16.
// ---------------------------------------------------------------------------
__global__ __launch_bounds__(256) void gemm2_kernel(
    const float* __restrict__ X, const float* __restrict__ W0,
    const float* __restrict__ W1, float* __restrict__ Y0,
    float* __restrict__ Y1, int strips) {
  __shared__ float tile[16 * LDSS];
  const int lane = threadIdx.x & 31;
  const int wave = threadIdx.x >> 5;  // n-tile 0..7
  const int nc = (wave << 4) + (lane & 15);  // output column
  const int koff = (lane >> 4) << 1;         // K sub-offset per half-wave
  const int arow = lane & 15;                // A row within tile

  // B fragments for this wave's 16 columns (persist in registers)
  v2f bfrag0[32], bfrag1[32];
#pragma unroll
  for (int ks = 0; ks < 32; ++ks) {
    const int k = (ks << 2) + koff;
    bfrag0[ks].x = W0[(size_t)k * D + nc];
    bfrag0[ks].y = W0[(size_t)(k + 1) * D + nc];
    bfrag1[ks].x = W1[(size_t)k * D + nc];
    bfrag1[ks].y = W1[(size_t)(k + 1) * D + nc];
  }

  for (int strip = blockIdx.x; strip < strips; strip += gridDim.x) {
    const size_t row0 = (size_t)strip * 16;
    // Cooperative coalesced load of the 16x128 A tile into LDS
    const float4* srcp = reinterpret_cast<const float4*>(X + row0 * D);
#pragma unroll
    for (int i = 0; i < 2; ++i) {
      const int idx = i * 256 + threadIdx.x;  // 0..511 float4s
      const float4 v = srcp[idx];
      const int flat = idx << 2;
      const int r = flat >> 7, c = flat & (D - 1);
      float* p = &tile[r * LDSS + c];
      p[0] = v.x; p[1] = v.y; p[2] = v.z; p[3] = v.w;
    }
    __syncthreads();

    v8f acc0 = {0.f, 0.f, 0.f, 0.f, 0.f, 0.f, 0.f, 0.f};
    v8f acc1 = {0.f, 0.f, 0.f, 0.f, 0.f, 0.f, 0.f, 0.f};
#pragma unroll
    for (int ks = 0; ks < 32; ++ks) {
      const int k = (ks << 2) + koff;
      v2f a;
      a.x = tile[arow * LDSS + k];
      a.y = tile[arow * LDSS + k + 1];
      acc0 = wmma_f32_4(a, bfrag0[ks], acc0);
      acc1 = wmma_f32_4(a, bfrag1[ks], acc1);
    }
    __syncthreads();  // tile reused next iteration

    // C layout: vgpr i -> row (i + 8*(lane>>4)), col = lane&15
    const size_t mb = row0 + (size_t)((lane >> 4) << 3);
#pragma unroll
    for (int i = 0; i < 8; ++i) {
      Y0[(mb + i) * D + nc] = acc0[i];
      Y1[(mb + i) * D + nc] = acc1[i];
    }
  }
}

// ---------------------------------------------------------------------------
// out (+)= tanh(X@W + bias); optional per-row scale X[r]/max(cnt[r],1).
// ---------------------------------------------------------------------------
__global__ __launch_bounds__(256) void gemm_out_kernel(
    const float* __restrict__ X, const float* __restrict__ W,
    const float* __restrict__ bias, const float* __restrict__ cntp,
    float* __restrict__ out, int accumulate, int strips) {
  __shared__ float tile[16 * LDSS];
  const int lane = threadIdx.x & 31;
  const int wave = threadIdx.x >> 5;
  const int nc = (wave << 4) + (lane & 15);
  const int koff = (lane >> 4) << 1;
  const int arow = lane & 15;

  v2f bfrag[32];
#pragma unroll
  for (int ks = 0; ks < 32; ++ks) {
    const int k = (ks << 2) + koff;
    bfrag[ks].x = W[(size_t)k * D + nc];
    bfrag[ks].y = W[(size_t)(k + 1) * D + nc];
  }
  const float bv = bias[nc];

  for (int strip = blockIdx.x; strip < strips; strip += gridDim.x) {
    const size_t row0 = (size_t)strip * 16;
    const float4* srcp = reinterpret_cast<const float4*>(X + row0 * D);
#pragma unroll
    for (int i = 0; i < 2; ++i) {
      const int idx = i * 256 + threadIdx.x;
      const float4 v = srcp[idx];
      const int flat = idx << 2;
      const int r = flat >> 7, c = flat & (D - 1);
      float sc = 1.0f;
      if (cntp) sc = 1.0f / fmaxf(cntp[row0 + r], 1.0f);
      float* p = &tile[r * LDSS + c];
      p[0] = v.x * sc; p[1] = v.y * sc; p[2] = v.z * sc; p[3] = v.w * sc;
    }
    __syncthreads();

    v8f acc = {0.f, 0.f, 0.f, 0.f, 0.f, 0.f, 0.f, 0.f};
#pragma unroll
    for (int ks = 0; ks < 32; ++ks) {
      const int k = (ks << 2) + koff;
      v2f a;
      a.x = tile[arow * LDSS + k];
      a.y = tile[arow * LDSS + k + 1];
      acc = wmma_f32_4(a, bfrag[ks], acc);
    }
    __syncthreads();

    const size_t mb = row0 + (size_t)((lane >> 4) << 3);
#pragma unroll
    for (int i = 0; i < 8; ++i) {
      const float val = tanhf(acc[i] + bv);
      const size_t p = (mb + i) * D + nc;
      out[p] = (accumulate ? out[p] : 0.0f) + val;
    }
  }
}

// ---------------------------------------------------------------------------
// Per-edge attention score: e = tanh(projA[idxA]+projB[idxB]+bias)·w + wb[0];
// atomicMax of positive scores into segmax[gid] (init 0 == max(0, segmax)).
// One wave per edge, 4 features per lane.
// ---------------------------------------------------------------------------
__global__ __launch_bounds__(256) void edge_score_kernel(
    const float* __restrict__ projA, const float* __restrict__ projB,
    const int* __restrict__ idxA, const int* __restrict__ idxB,
    const int* __restrict__ gid, const float* __restrict__ bias,
    const float* __restrict__ w, const float* __restrict__ wb,
    float* __restrict__ escore, float* __restrict__ segmax, int nedges) {
  const int e = blockIdx.x * 8 + (threadIdx.x >> 5);
  const int lane = threadIdx.x & 31;
  if (e >= nedges) return;
  const int ia = idxA[e], ib = idxB[e];
  const float4 va = ((const float4*)(projA + (size_t)ia * D))[lane];
  const float4 vb = ((const float4*)(projB + (size_t)ib * D))[lane];
  const float4 vbi = ((const float4*)bias)[lane];
  const float4 vw = ((const float4*)w)[lane];
  float s = tanhf(va.x + vb.x + vbi.x) * vw.x +
            tanhf(va.y + vb.y + vbi.y) * vw.y +
            tanhf(va.z + vb.z + vbi.z) * vw.z +
            tanhf(va.w + vb.w + vbi.w) * vw.w;
  for (int off = 16; off > 0; off >>= 1) s += __shfl_xor(s, off, 32);
  if (lane == 0) {
    const float val = s + wb[0];
    escore[e] = val;
    // int-bit atomicMax valid: running max stays >= 0 (memset init),
    // and positive float bits are monotonic as ints.
    if (val > 0.0f) atomicMax((int*)(segmax + gid[e]), __float_as_int(val));
  }
}

__global__ __launch_bounds__(256) void edge_exp_kernel(
    float* __restrict__ escore, const float* __restrict__ segmax,
    const int* __restrict__ gid, float* __restrict__ segsum, int nedges) {
  const int e = blockIdx.x * 256 + threadIdx.x;
  if (e >= nedges) return;
  const int g = gid[e];
  const float ex = expf(escore[e] - segmax[g]);
  escore[e] = ex;
  atomicAdd(segsum + g, ex);
}

// agg[gid] += (ex / (segsum[gid]+1e-9)) * emb[vidx]; one wave per edge.
__global__ __launch_bounds__(256) void edge_agg_kernel(
    const float* __restrict__ escore, const float* __restrict__ segsum,
    const int* __restrict__ gid, const int* __restrict__ vidx,
    const float* __restrict__ emb, float* __restrict__ agg, int nedges) {
  const int e = blockIdx.x * 8 + (threadIdx.x >> 5);
  const int lane = threadIdx.x & 31;
  if (e >= nedges) return;
  const int g = gid[e], v = vidx[e];
  const float attn = escore[e] / (segsum[g] + 1e-9f);
  const float4 r = ((const float4*)(emb + (size_t)v * D))[lane];
  float* dst = agg + (size_t)g * D + lane * 4;
  atomicAdd(dst + 0, attn * r.x);
  atomicAdd(dst + 1, attn * r.y);
  atomicAdd(dst + 2, attn * r.z);
  atomicAdd(dst + 3, attn * r.w);
}

// mean[src] += emb[dst]; cnt[src] += 1. One wave per edge.
__global__ __launch_bounds__(256) void mean_edge_kernel(
    const int* __restrict__ srci, const int* __restrict__ dsti,
    const float* __restrict__ emb, float* __restrict__ meanb,
    float* __restrict__ cntp, int nedges) {
  const int e = blockIdx.x * 8 + (threadIdx.x >> 5);
  const int lane = threadIdx.x & 31;
  if (e >= nedges) return;
  const int sn = srci[e], dn = dsti[e];
  const float4 r = ((const float4*)(emb + (size_t)dn * D))[lane];
  float* q = meanb + (size_t)sn * D + lane * 4;
  atomicAdd(q + 0, r.x);
  atomicAdd(q + 1, r.y);
  atomicAdd(q + 2, r.z);
  atomicAdd(q + 3, r.w);
  if (lane == 0) atomicAdd(cntp + sn, 1.0f);
}

extern "C" void kernel_launch(void* const* d_in, const int* in_sizes, int n_in,
                              void* d_out, int out_size, void* d_ws,
                              size_t ws_size, hipStream_t stream) {
  (void)n_in; (void)out_size; (void)ws_size;
  const float* emb  = (const float*)d_in[0];
  const float* Wae  = (const float*)d_in[1];
  const float* bae  = (const float*)d_in[2];
  const float* w0w  = (const float*)d_in[3];
  const float* w0b  = (const float*)d_in[4];
  const float* War  = (const float*)d_in[5];
  const float* bar  = (const float*)d_in[6];
  const float* w1w  = (const float*)d_in[7];
  const float* w1b  = (const float*)d_in[8];
  const float* W1   = (const float*)d_in[9];
  const float* b1   = (const float*)d_in[10];
  const float* W2   = (const float*)d_in[11];
  const float* b2   = (const float*)d_in[12];
  const float* W3   = (const float*)d_in[13];
  const float* b3   = (const float*)d_in[14];
  const int* er_src = (const int*)d_in[15];
  const int* er_dst = (const int*)d_in[16];
  const int* ee_src = (const int*)d_in[17];
  const int* ee_dst = (const int*)d_in[18];
  const int* rr_src = (const int*)d_in[19];
  const int* rr_dst = (const int*)d_in[20];

  const int N   = in_sizes[0] / D;   // 100000
  const int Eer = in_sizes[15];      // 500000
  const int Eee = in_sizes[17];      // 500000
  const int Err = in_sizes[19];      // 200000

  float* ws = (float*)d_ws;
  const size_t nf = (size_t)N * D;
  float* projA    = ws;               // N*128
  float* projB    = projA + nf;       // N*128
  float* agg      = projB + nf;       // N*128 attention accumulator
  float* esc0     = agg + nf;         // Eer floats
  float* esc1     = esc0 + Eer;       // Eer floats
  float* segmax0  = esc1 + Eer;       // N floats (segmax0..cntbuf contiguous)
  float* segsum0  = segmax0 + N;      // N floats
  float* segmax1  = segsum0 + N;      // N floats
  float* segsum1  = segmax1 + N;      // N floats
  float* cntbuf   = segsum1 + N;      // N floats
  float* meanb    = projA;            // reused after attention scoring
  float* out      = (float*)d_out;

  const int strips = N / 16;          // N divisible by 16 for this problem
  const int gemm_blocks = strips < 2048 ? strips : 2048;
  dim3 blk(256);
  dim3 gG(gemm_blocks);
  dim3 gE((Eer + 7) / 8), gE1((Eer + 255) / 256);

  (void)hipMemsetAsync(agg, 0, nf * sizeof(float), stream);
  (void)hipMemsetAsync(segmax0, 0, (size_t)5 * N * sizeof(float), stream);

  // ---- entity-side attention (grouped by er_src, values r = emb[er_dst]) ----
  gemm2_kernel<<<gG, blk, 0, stream>>>(emb, Wae, Wae + D * D, projA, projB,
                                       strips);
  edge_score_kernel<<<gE, blk, 0, stream>>>(projA, projB, er_dst, er_src,
                                            er_src, bae, w0w, w0b, esc0,
                                            segmax0, Eer);
  edge_exp_kernel<<<gE1, blk, 0, stream>>>(esc0, segmax0, er_src, segsum0, Eer);
  edge_agg_kernel<<<gE, blk, 0, stream>>>(esc0, segsum0, er_src, er_dst, emb,
                                          agg, Eer);

  // ---- relation-side attention (grouped by er_dst, values h = emb[er_src]) --
  gemm2_kernel<<<gG, blk, 0, stream>>>(emb, War, War + D * D, projA, projB,
                                       strips);
  edge_score_kernel<<<gE, blk, 0, stream>>>(projA, projB, er_src, er_dst,
                                            er_dst, bar, w1w, w1b, esc1,
                                            segmax1, Eer);
  edge_exp_kernel<<<gE1, blk, 0, stream>>>(esc1, segmax1, er_dst, segsum1, Eer);
  edge_agg_kernel<<<gE, blk, 0, stream>>>(esc1, segsum1, er_dst, er_src, emb,
                                          agg, Eer);

  // ---- mean aggregation (reuses projA storage) ----
  (void)hipMemsetAsync(meanb, 0, nf * sizeof(float), stream);
  dim3 gM1((Eee + 7) / 8), gM2((Err + 7) / 8);
  mean_edge_kernel<<<gM1, blk, 0, stream>>>(ee_src, ee_dst, emb, meanb, cntbuf,
                                            Eee);
  mean_edge_kernel<<<gM2, blk, 0, stream>>>(rr_src, rr_dst, emb, meanb, cntbuf,
                                            Err);

  // ---- final: tanh(emb@W1+b1) + tanh(agg@W2+b2) + tanh(mean@W3+b3) ----
  gemm_out_kernel<<<gG, blk, 0, stream>>>(emb, W1, b1, nullptr, out, 0, strips);
  gemm_out_kernel<<<gG, blk, 0, stream>>>(agg, W2, b2, nullptr, out, 1, strips);
  gemm_out_kernel<<<gG, blk, 0, stream>>>(meanb, W3, b3, cntbuf, out, 1,
                                          strips);
}